// GCNEncoder_78357383348247
// MI455X (gfx1250) — compile-verified
//
#include <hip/hip_runtime.h>
#include <hip/hip_bf16.h>

typedef __attribute__((ext_vector_type(2))) float v2f;
typedef __attribute__((ext_vector_type(8))) float v8f;

#define GCN_N 100000
#define GCN_E 1600000

// ---------------------------------------------------------------------------
// Degree / norm kernels
// ---------------------------------------------------------------------------
__global__ void init_deg_kernel(float* __restrict__ deg, int n) {
    int i = blockIdx.x * blockDim.x + threadIdx.x;
    if (i < n) deg[i] = 1.0f;  // self-loop contributes 1
}

__global__ void edge_deg_kernel(const long long* __restrict__ ei,
                                float* __restrict__ deg, int e) {
    int i = blockIdx.x * blockDim.x + threadIdx.x;
    if (i < e) {
        int col = (int)ei[GCN_E + i];  // target node
        atomicAdd(&deg[col], 1.0f);
    }
}

__global__ void dinv_kernel(const float* __restrict__ deg,
                            float* __restrict__ dinv, int n) {
    int i = blockIdx.x * blockDim.x + threadIdx.x;
    if (i < n) dinv[i] = __frsqrt_rn(deg[i]);  // deg >= 1 always (self loops)
}

// ---------------------------------------------------------------------------
// FP32 WMMA GEMM:  out[N x NOUT] = A[N x K] * W[K x NOUT]
// One wave (32 threads) computes one 16x16 output tile via V_WMMA_F32_16X16X4_F32.
//
// A-matrix 16x4 f32 layout (ISA 7.12.2): lanes 0-15 rows M=0..15 with K={0,1}
// in VGPR0/1; lanes 16-31 same rows with K={2,3}.
// B-matrix 4x16: lane = output column, same K striping.
// C/D 16x16 f32: VGPR r, lane L -> (M = r + 8*(L>>4), N = L&15).
// ---------------------------------------------------------------------------
template <int K, int NOUT>
__global__ __launch_bounds__(32) void gemm_f32_wmma_kernel(
    const float* __restrict__ A, const float* __restrict__ W,
    float* __restrict__ out) {
    const int m0   = blockIdx.x * 16;
    const int n0   = blockIdx.y * 16;
    const int lane = threadIdx.x;
    const int half = lane >> 4;   // 0: K+{0,1}, 1: K+{2,3}
    const int l15  = lane & 15;
    const int koff = half * 2;

    const float* __restrict__ arow = A + (size_t)(m0 + l15) * K;
    const float* __restrict__ wcol = W + (n0 + l15);

    v8f acc = {};
#pragma unroll
    for (int k = 0; k < K; k += 4) {
        v2f a, b;
        a.x = arow[k + koff];
        a.y = arow[k + koff + 1];
        b.x = wcol[(size_t)(k + koff) * NOUT];
        b.y = wcol[(size_t)(k + koff + 1) * NOUT];
        acc = __builtin_amdgcn_wmma_f32_16x16x4_f32(
            /*neg_a=*/false, a, /*neg_b=*/false, b,
            /*c_mod=*/(short)0, acc, /*reuse_a=*/false, /*reuse_b=*/false);
    }

#pragma unroll
    for (int r = 0; r < 8; ++r) {
        out[(size_t)(m0 + r + 8 * half) * NOUT + n0 + l15] = acc[r];
    }
}

// ---------------------------------------------------------------------------
// Aggregation: init with self-loop term (+bias folded), edge scatter atomics,
// then elementwise activation.
// ---------------------------------------------------------------------------
template <int F>
__global__ void init_agg_kernel(const float* __restrict__ hW,
                                const float* __restrict__ dinv,
                                const float* __restrict__ bias,
                                float* __restrict__ agg, int total) {
    int i = blockIdx.x * blockDim.x + threadIdx.x;
    if (i < total) {
        int node = i / F;
        int f    = i % F;
        float dv = dinv[node];
        agg[i] = hW[i] * dv * dv + bias[f];
    }
}

// One thread per (edge, 4-feature chunk). F/4 chunks per edge.
template <int F>
__global__ void edge_scatter_kernel(const long long* __restrict__ ei,
                                    const float* __restrict__ hW,
                                    const float* __restrict__ dinv,
                                    float* __restrict__ agg, int total) {
    constexpr int CH = F / 4;
    int i = blockIdx.x * blockDim.x + threadIdx.x;
    if (i >= total) return;
    int e = i / CH;
    int c = i % CH;
    int r  = (int)ei[e];           // source
    int cl = (int)ei[GCN_E + e];   // target
    float w = dinv[r] * dinv[cl];
    const float4 v = *(const float4*)(hW + (size_t)r * F + c * 4);
    float* dst = agg + (size_t)cl * F + c * 4;
    atomicAdd(dst + 0, v.x * w);
    atomicAdd(dst + 1, v.y * w);
    atomicAdd(dst + 2, v.z * w);
    atomicAdd(dst + 3, v.w * w);
}

__global__ void relu_inplace_kernel(float* __restrict__ a, int total) {
    int i = blockIdx.x * blockDim.x + threadIdx.x;
    if (i < total) a[i] = fmaxf(a[i], 0.0f);
}

// ---------------------------------------------------------------------------
// Launch
// ---------------------------------------------------------------------------
extern "C" void kernel_launch(void* const* d_in, const int* in_sizes, int n_in,
                              void* d_out, int out_size, void* d_ws, size_t ws_size,
                              hipStream_t stream) {
    const float*     x  = (const float*)d_in[0];       // [N,128]
    const long long* ei = (const long long*)d_in[1];   // [2,E] int64
    const float*     W1 = (const float*)d_in[2];       // [128,64]
    const float*     b1 = (const float*)d_in[3];       // [64]
    const float*     W2 = (const float*)d_in[4];       // [64,32]
    const float*     b2 = (const float*)d_in[5];       // [32]
    float*           out = (float*)d_out;              // [N,32]

    constexpr int N = GCN_N, E = GCN_E;
    constexpr int HID = 64, OUT_F = 32, IN_F = 128;

    float* ws   = (float*)d_ws;
    float* deg  = ws;                           // N
    float* dinv = ws + N;                       // N
    float* xW   = ws + 2 * (size_t)N;           // N*64
    float* agg1 = xW + (size_t)N * HID;         // N*64 (becomes h1 after relu)
    float* h2W  = agg1 + (size_t)N * HID;       // N*32

    const int B = 256;

    // 1) degree + normalization
    init_deg_kernel<<<(N + B - 1) / B, B, 0, stream>>>(deg, N);
    edge_deg_kernel<<<(E + B - 1) / B, B, 0, stream>>>(ei, deg, E);
    dinv_kernel<<<(N + B - 1) / B, B, 0, stream>>>(deg, dinv, N);

    // 2) xW = x @ W1  (WMMA fp32)
    {
        dim3 grid(N / 16, HID / 16);
        gemm_f32_wmma_kernel<IN_F, HID><<<grid, 32, 0, stream>>>(x, W1, xW);
    }

    // 3) layer-1 aggregation: self loops (+b1), edge atomics, relu
    init_agg_kernel<HID><<<(N * HID + B - 1) / B, B, 0, stream>>>(
        xW, dinv, b1, agg1, N * HID);
    {
        int total = E * (HID / 4);
        edge_scatter_kernel<HID><<<(total + B - 1) / B, B, 0, stream>>>(
            ei, xW, dinv, agg1, total);
    }
    relu_inplace_kernel<<<(N * HID + B - 1) / B, B, 0, stream>>>(agg1, N * HID);

    // 4) h2W = h1 @ W2  (WMMA fp32)
    {
        dim3 grid(N / 16, OUT_F / 16);
        gemm_f32_wmma_kernel<HID, OUT_F><<<grid, 32, 0, stream>>>(agg1, W2, h2W);
    }

    // 5) layer-2 aggregation directly into d_out: self loops (+b2), edge atomics
    init_agg_kernel<OUT_F><<<(N * OUT_F + B - 1) / B, B, 0, stream>>>(
        h2W, dinv, b2, out, N * OUT_F);
    {
        int total = E * (OUT_F / 4);
        edge_scatter_kernel<OUT_F><<<(total + B - 1) / B, B, 0, stream>>>(
            ei, h2W, dinv, out, total);
    }
}